// Diversity5_52699248721937
// MI455X (gfx1250) — compile-verified
//
#include <hip/hip_runtime.h>
#include <math.h>

// Diversity5: mean over B of SCALE * sum_{m<n} corr(softmax(x_m/T), softmax(x_n/T))
// corr per row computed as V*V^T via V_WMMA_F32_16X16X4_F32 (symmetric: A regs == B regs).

typedef __attribute__((ext_vector_type(2))) float v2f;
typedef __attribute__((ext_vector_type(8))) float v8f;

#define C_CLASSES 1000
#define B_ROWS    16384
#define INV_T     (1.0f / 20.0f)
#define SCALE_F   0.3f

__device__ __forceinline__ float wave_reduce_add(float v) {
#pragma unroll
    for (int off = 16; off > 0; off >>= 1)
        v += __shfl_xor(v, off, 32);
    return v;
}

__global__ __launch_bounds__(256)
void div5_corr_kernel(const float* __restrict__ x0, const float* __restrict__ x1,
                      const float* __restrict__ x2, const float* __restrict__ x3,
                      const float* __restrict__ x4, float* __restrict__ ws) {
    const int lane        = threadIdx.x & 31;
    const int waveInBlock = threadIdx.x >> 5;
    const int wavesPerBlk = blockDim.x >> 5;
    const int waveId      = blockIdx.x * wavesPerBlk + waveInBlock;
    const int nWaves      = gridDim.x * wavesPerBlk;

    const float* xs[5] = {x0, x1, x2, x3, x4};

    // Per-lane WMMA operand roles (16x16x4 f32):
    //   A: lanes 0-15 -> M=lane, VGPR0=K0 / VGPR1=K1; lanes 16-31 -> M=lane-16, K2/K3
    //   B: lanes 0-15 -> N=lane, VGPR0=K0 / VGPR1=K1; lanes 16-31 -> N=lane-16, K2/K3
    // => for symmetric V*V^T, A and B register images coincide.
    const int M    = lane & 15;          // matrix row this lane supplies (valid if < 5)
    const int koff = (lane >> 4) << 1;   // 0 for lanes 0-15, 2 for lanes 16-31

    // loop-invariant source pointer for this lane's matrix row
    const float* pm = (M == 1) ? x1 : (M == 2) ? x2 : (M == 3) ? x3
                    : (M == 4) ? x4 : x0;

    // loop-invariant upper-triangle weights: pair (m, n=lane) counted iff m < n <= 4.
    // C-tile layout: acc[j], lanes 0-15 hold D[M=j][N=lane].
    const float w0 = (lane >= 1 && lane <= 4) ? 1.0f : 0.0f;
    const float w1 = (lane >= 2 && lane <= 4) ? 1.0f : 0.0f;
    const float w2 = (lane >= 3 && lane <= 4) ? 1.0f : 0.0f;
    const float w3 = (lane == 4) ? 1.0f : 0.0f;

    float waveAcc = 0.0f;

    for (int b = waveId; b < B_ROWS; b += nWaves) {
        // ---------- pass 1: softmax stats per model: Z = sum e, Q = sum e^2 ----------
        // 128-bit streaming loads: 250 float4 per row = 7 full wave iters + 26-lane tail.
        float am = 0.0f, gm = 0.0f;     // this lane's alpha/gamma (0 for M >= 5)
        const int bn = b + nWaves;      // prefetch target row
#pragma unroll
        for (int m = 0; m < 5; ++m) {
            const float4* row4 =
                reinterpret_cast<const float4*>(xs[m] + (size_t)b * C_CLASSES);
            if (bn < B_ROWS) {
                // pull the next row this wave will process into cache (global_prefetch_b8)
                __builtin_prefetch(xs[m] + (size_t)bn * C_CLASSES + lane * 32, 0, 3);
            }
            float Z = 0.0f, Q = 0.0f;
#pragma unroll
            for (int j = 0; j < 7; ++j) {
                float4 xv = row4[lane + 32 * j];
                float e0 = __expf(xv.x * INV_T);
                float e1 = __expf(xv.y * INV_T);
                float e2 = __expf(xv.z * INV_T);
                float e3 = __expf(xv.w * INV_T);
                Z += (e0 + e1) + (e2 + e3);
                Q += (e0 * e0 + e1 * e1) + (e2 * e2 + e3 * e3);
            }
            {   // tail: float4 indices 224..249 (26 lanes)
                int ti = 224 + lane;
                if (ti < C_CLASSES / 4) {
                    float4 xv = row4[ti];
                    float e0 = __expf(xv.x * INV_T);
                    float e1 = __expf(xv.y * INV_T);
                    float e2 = __expf(xv.z * INV_T);
                    float e3 = __expf(xv.w * INV_T);
                    Z += (e0 + e1) + (e2 + e3);
                    Q += (e0 * e0 + e1 * e1) + (e2 * e2 + e3 * e3);
                }
            }
            Z = wave_reduce_add(Z);
            Q = wave_reduce_add(Q);
            // mean of softmax = 1/C exactly; ||v||^2 = Q/Z^2 - 1/C
            float normSq = Q / (Z * Z) - (1.0f / (float)C_CLASSES);
            float rinv   = rsqrtf(normSq);
            float al     = rinv / Z;                          // alpha_m = 1/(Z*r)
            float ga     = rinv * (1.0f / (float)C_CLASSES);  // gamma_m = 1/(C*r)
            // branch-free deposit into this lane's coefficients (cndmask)
            if (M == m) { am = al; gm = ga; }
        }

        const float* rowm = pm + (size_t)b * C_CLASSES;

        // ---------- pass 2: corr = V * V^T accumulated in 16x16 f32 tile ----------
        v8f acc = {};
#pragma unroll 4
        for (int k0 = 0; k0 < C_CLASSES; k0 += 4) {
            const int c = k0 + koff;                    // c, c+1 in [0, 999]
            float e0 = __expf(rowm[c]     * INV_T);
            float e1 = __expf(rowm[c + 1] * INV_T);
            v2f a;
            a.x = am * e0 - gm;                          // v[M][c]   (0 for M>=5)
            a.y = am * e1 - gm;                          // v[M][c+1]
            acc = __builtin_amdgcn_wmma_f32_16x16x4_f32(
                false, a, false, a, (short)0, acc, false, false);
        }

        // ---------- branch-free upper-triangle sum from C tile ----------
        float s = w0 * acc[0] + w1 * acc[1] + w2 * acc[2] + w3 * acc[3];
        waveAcc += wave_reduce_add(s);
    }

    if (lane == 0) ws[waveId] = waveAcc;
}

__global__ __launch_bounds__(256)
void div5_finalize(const float* __restrict__ ws, float* __restrict__ out, int n) {
    __shared__ float sm[256];
    float s = 0.0f;
    for (int i = threadIdx.x; i < n; i += 256) s += ws[i];
    sm[threadIdx.x] = s;
    __syncthreads();
    for (int off = 128; off > 0; off >>= 1) {
        if ((int)threadIdx.x < off) sm[threadIdx.x] += sm[threadIdx.x + off];
        __syncthreads();
    }
    if (threadIdx.x == 0) out[0] = sm[0] * (SCALE_F / (float)B_ROWS);
}

extern "C" void kernel_launch(void* const* d_in, const int* in_sizes, int n_in,
                              void* d_out, int out_size, void* d_ws, size_t ws_size,
                              hipStream_t stream) {
    const float* x0 = (const float*)d_in[0];
    const float* x1 = (const float*)d_in[1];
    const float* x2 = (const float*)d_in[2];
    const float* x3 = (const float*)d_in[3];
    const float* x4 = (const float*)d_in[4];
    // d_in[5] (targets) is unused by the reference computation.
    float* ws  = (float*)d_ws;
    float* out = (float*)d_out;

    int blocks  = 256;                       // 256 blocks x 8 waves = 2048 waves
    const int threads = 256;
    // keep the per-wave partial array within the provided scratch
    size_t needWaves = (size_t)blocks * (threads / 32);
    if (ws_size < needWaves * sizeof(float)) {
        size_t maxWaves = ws_size / sizeof(float);
        int maxBlocks = (int)(maxWaves / (threads / 32));
        blocks = maxBlocks > 0 ? maxBlocks : 1;
    }
    const int nWaves = blocks * (threads / 32);

    div5_corr_kernel<<<blocks, threads, 0, stream>>>(x0, x1, x2, x3, x4, ws);
    div5_finalize<<<1, 256, 0, stream>>>(ws, out, nWaves);
}